// DeformationGraph_8083128451095
// MI455X (gfx1250) — compile-verified
//
#include <hip/hip_runtime.h>

typedef __attribute__((ext_vector_type(2))) float v2f;
typedef __attribute__((ext_vector_type(8))) float v8f;

#define B_      4
#define V_      50000
#define C_      160
#define KSTEPS  40        // C_/4
#define WAVES   8
#define NTILES  3125      // V_/16

__global__ __launch_bounds__(256) void deform_lbs_wmma(
    const float* __restrict__ X,      // (B,V,3)
    const float* __restrict__ Vn,     // (B,C,3)
    const float* __restrict__ rot6d,  // (B,C,6)
    const float* __restrict__ W,      // (V,C)
    const int*   __restrict__ idx,    // (C,)
    float* __restrict__ out)          // (B,V,3)
{
    // B-matrix fragments for all 4 batches, pre-swizzled to WMMA lane layout.
    // [b][kk][lane][2 comps] : rows (c=4kk+r): r in {0,1} -> lanes 0-15 comp r,
    //                          r in {2,3} -> lanes 16-31 comp r-2.
    __shared__ __align__(16) float s_fragB[B_ * KSTEPS * 32 * 2];  // 40 KB
    __shared__ __align__(16) float s_g[WAVES * 256];               // 8 KB epilogue scratch

    const int tid = threadIdx.x;

    // ---------------- Phase 0: build per-(b,c) affine M = [R | t] ----------------
    for (int t = tid; t < B_ * C_; t += 256) {
        const int b = t / C_;
        const int c = t % C_;

        const float* d6 = rot6d + ((size_t)b * C_ + c) * 6;
        float a1x = d6[0], a1y = d6[1], a1z = d6[2];
        float a2x = d6[3], a2y = d6[4], a2z = d6[5];
        const float eps = 1e-8f;

        float n1 = fmaxf(sqrtf(a1x*a1x + a1y*a1y + a1z*a1z), eps);
        float b1x = a1x / n1, b1y = a1y / n1, b1z = a1z / n1;
        float dp = b1x*a2x + b1y*a2y + b1z*a2z;
        float px = a2x - dp*b1x, py = a2y - dp*b1y, pz = a2z - dp*b1z;
        float n2 = fmaxf(sqrtf(px*px + py*py + pz*pz), eps);
        float b2x = px / n2, b2y = py / n2, b2z = pz / n2;
        float b3x = b1y*b2z - b1z*b2y;
        float b3y = b1z*b2x - b1x*b2z;
        float b3z = b1x*b2y - b1y*b2x;

        const int vi = idx[c];
        const float* cp = X + ((size_t)b * V_ + vi) * 3;
        float c0 = cp[0], c1 = cp[1], c2 = cp[2];
        const float* vp = Vn + ((size_t)b * C_ + c) * 3;

        float t0 = c0 + vp[0] - (b1x*c0 + b1y*c1 + b1z*c2);
        float t1 = c1 + vp[1] - (b2x*c0 + b2y*c1 + b2z*c2);
        float t2 = c2 + vp[2] - (b3x*c0 + b3y*c1 + b3z*c2);

        float m[16] = { b1x, b1y, b1z, t0,
                        b2x, b2y, b2z, t1,
                        b3x, b3y, b3z, t2,
                        0.f, 0.f, 0.f, 0.f };

        const int kk       = c >> 2;
        const int r        = c & 3;
        const int laneBase = (r >= 2) ? 16 : 0;
        const int comp     = r & 1;
        float* dst = &s_fragB[(((b * KSTEPS) + kk) * 32 + laneBase) * 2 + comp];
        #pragma unroll
        for (int n = 0; n < 16; ++n)
            dst[n * 2] = m[n];
    }
    __syncthreads();

    // ---------------- Phase 1: WMMA GEMM  G[b] = W_tile(16x160) * M[b](160x16) ----
    const int wave = tid >> 5;
    const int lane = tid & 31;
    const int tile = blockIdx.x * WAVES + wave;
    const bool valid = (tile < NTILES);
    const int vbase = valid ? tile * 16 : 0;
    const int mrow  = lane & 15;
    const int colBase = (lane < 16) ? 0 : 2;                // A-frag K-pair select
    const float* wrow = W + (size_t)(vbase + mrow) * C_ + colBase;

    v8f acc[B_];
    #pragma unroll
    for (int b = 0; b < B_; ++b) acc[b] = (v8f){0,0,0,0,0,0,0,0};

    for (int kk = 0; kk < KSTEPS; ++kk) {
        v2f a = *reinterpret_cast<const v2f*>(wrow + 4 * kk);   // global_load_b64
        #pragma unroll
        for (int b = 0; b < B_; ++b) {
            v2f bf = *reinterpret_cast<const v2f*>(
                &s_fragB[(((b * KSTEPS) + kk) * 32 + lane) * 2]); // ds_load_b64
            acc[b] = __builtin_amdgcn_wmma_f32_16x16x4_f32(
                /*neg_a=*/false, a, /*neg_b=*/false, bf,
                /*c_mod=*/(short)0, acc[b], /*reuse_a=*/false, /*reuse_b=*/false);
        }
    }

    // ---------------- Phase 2: epilogue — apply blended affine to [X,1] ----------
    float* g = &s_g[wave * 256];
    #pragma unroll
    for (int b = 0; b < B_; ++b) {
        const int rowOff = (lane < 16) ? 0 : 8;
        const int n = lane & 15;
        #pragma unroll
        for (int r = 0; r < 8; ++r)
            g[(rowOff + r) * 16 + n] = acc[b][r];
        __syncthreads();
        if (lane < 16) {
            const int vv = vbase + lane;
            const float* gr = &g[lane * 16];
            const float* xp = X + ((size_t)b * V_ + vv) * 3;
            float x = xp[0], y = xp[1], z = xp[2];
            float o0 = gr[0]*x + gr[1]*y + gr[2]*z  + gr[3];
            float o1 = gr[4]*x + gr[5]*y + gr[6]*z  + gr[7];
            float o2 = gr[8]*x + gr[9]*y + gr[10]*z + gr[11];
            if (valid) {
                float* op = out + ((size_t)b * V_ + vv) * 3;
                op[0] = o0; op[1] = o1; op[2] = o2;
            }
        }
        __syncthreads();
    }
}

extern "C" void kernel_launch(void* const* d_in, const int* in_sizes, int n_in,
                              void* d_out, int out_size, void* d_ws, size_t ws_size,
                              hipStream_t stream) {
    const float* X   = (const float*)d_in[0];
    const float* Vn  = (const float*)d_in[1];
    const float* r6  = (const float*)d_in[2];
    const float* W   = (const float*)d_in[3];
    const int*   idx = (const int*)d_in[4];
    float* out = (float*)d_out;

    dim3 grid((NTILES + WAVES - 1) / WAVES);  // 391 blocks x 256 threads (8 waves)
    deform_lbs_wmma<<<grid, 256, 0, stream>>>(X, Vn, r6, W, idx, out);
}